// DiffFDN_59219009077318
// MI455X (gfx1250) — compile-verified
//
#include <hip/hip_runtime.h>
#include <hip/hip_bf16.h>

#define N_LINES 16
#define IR_LEN 48000
#define BATCH 32
#define SEQ 512
#define FEAT 256
#define BUF_TOTAL 29010   // sum of delay lengths
#define FSPLIT 4          // seq-axis split for feat reduction

static __device__ const int c_delays[16] = {1009,1123,1231,1321,1433,1543,1657,1777,
                                            1879,1987,2081,2179,2287,2383,2503,2617};
static __device__ const int c_off[16]    = {0,1009,2132,3363,4684,6117,7660,9317,
                                            11094,12973,14960,17041,19220,21507,23890,26393};

typedef float v2f __attribute__((ext_vector_type(2)));
typedef float v8f __attribute__((ext_vector_type(8)));

// -------- Phase 1: partial sums of x over a SEQ slice (grid: BATCH x FSPLIT) --------
__global__ void feat_kernel(const float* __restrict__ x, float* __restrict__ fpart) {
    int b = blockIdx.x, g = blockIdx.y, f = threadIdx.x;
    const int rows = SEQ / FSPLIT;                       // 128
    const float* xb = x + (size_t)b * SEQ * FEAT + (size_t)g * rows * FEAT + f;
    float s = 0.f;
    #pragma unroll 8
    for (int t = 0; t < rows; ++t) s += xb[t * FEAT];
    fpart[((size_t)b * FSPLIT + g) * FEAT + f] = s;
}

// -------- Phase 2: heads (tanh linear), skew, expm (scale&square + Taylor-12), A_g --------
__global__ void head_expm_kernel(const float* __restrict__ fpart,
                                 const float* __restrict__ WA, const float* __restrict__ bA,
                                 const float* __restrict__ WB, const float* __restrict__ bB,
                                 const float* __restrict__ WC, const float* __restrict__ bC,
                                 float* __restrict__ Ag, float* __restrict__ Bv,
                                 float* __restrict__ Cv) {
    __shared__ float f[FEAT];
    __shared__ float S[256], T[256], P[256], Acc[256];
    __shared__ float red[16];
    __shared__ float s_scale;
    __shared__ int   s_squar;

    int b = blockIdx.x, tid = threadIdx.x;
    int i = tid >> 4, j = tid & 15;

    // fold the FSPLIT partial sums -> mean
    {
        const float* p = fpart + (size_t)b * FSPLIT * FEAT + tid;
        float s = 0.f;
        #pragma unroll
        for (int g = 0; g < FSPLIT; ++g) s += p[g * FEAT];
        f[tid] = s * (1.0f / (float)SEQ);
    }
    __syncthreads();

    // A head: Ahat = tanh(feat @ WA^T + bA) -> stash in S
    {
        const float* w = WA + (size_t)tid * FEAT;
        float acc = bA[tid];
        for (int k = 0; k < FEAT; ++k) acc += f[k] * w[k];
        S[tid] = tanhf(acc);
    }
    if (tid < 16) {
        const float* w = WB + (size_t)tid * FEAT;
        float acc = bB[tid];
        for (int k = 0; k < FEAT; ++k) acc += f[k] * w[k];
        Bv[b * 16 + tid] = tanhf(acc);
    } else if (tid < 32) {
        int r = tid - 16;
        const float* w = WC + (size_t)r * FEAT;
        float acc = bC[r];
        for (int k = 0; k < FEAT; ++k) acc += f[k] * w[k];
        Cv[b * 16 + r] = tanhf(acc);
    }
    __syncthreads();

    // skew: S = triu(A,1) - triu(A,1)^T
    float sk = (i < j) ? S[i * 16 + j] : ((i > j) ? -S[j * 16 + i] : 0.f);
    __syncthreads();
    S[tid] = sk;
    __syncthreads();

    // inf-norm -> scaling
    if (j == 0) {
        float rs = 0.f;
        for (int k = 0; k < 16; ++k) rs += fabsf(S[i * 16 + k]);
        red[i] = rs;
    }
    __syncthreads();
    if (tid == 0) {
        float nm = 0.f;
        for (int k = 0; k < 16; ++k) nm = fmaxf(nm, red[k]);
        int sq = 0; float sc = 1.f;
        while (nm > 0.25f && sq < 10) { nm *= 0.5f; sc *= 0.5f; ++sq; }
        s_scale = sc; s_squar = sq;
    }
    __syncthreads();

    T[tid]   = S[tid] * s_scale;
    P[tid]   = (i == j) ? 1.f : 0.f;
    Acc[tid] = P[tid];
    __syncthreads();

    // Taylor order 12: Acc = sum_k T^k / k!
    for (int k = 1; k <= 12; ++k) {
        float t = 0.f;
        for (int m = 0; m < 16; ++m) t += P[i * 16 + m] * T[m * 16 + j];
        t *= (1.0f / (float)k);
        __syncthreads();
        P[tid] = t; Acc[tid] += t;
        __syncthreads();
    }
    // squaring
    int sq = s_squar;
    for (int q = 0; q < sq; ++q) {
        float t = 0.f;
        for (int m = 0; m < 16; ++m) t += Acc[i * 16 + m] * Acc[m * 16 + j];
        __syncthreads();
        Acc[tid] = t;
        __syncthreads();
    }
    // A_g = expm(S) @ diag(G), G[j] = 10^(-3*delay[j]/48000)
    float G = powf(10.0f, (-3.0f / 48000.0f) * (float)c_delays[j]);
    Ag[(size_t)b * 256 + tid] = Acc[tid] * G;
}

// -------- Phase 3: FDN recurrence. 32 waves/block, one 16-step tile per wave per
// 512-step chunk => minimal sequential depth (94 dependent iterations). Delay lines
// resident in LDS; 16x16 matvec batches on the f32 WMMA pipe. --------
__launch_bounds__(1024)
__global__ void fdn_kernel(const float* __restrict__ Ag,
                           const float* __restrict__ Bv,
                           const float* __restrict__ Cv,
                           float* __restrict__ y) {
    __shared__ float buf[BUF_TOTAL];
    int b = blockIdx.x, tid = threadIdx.x;
    for (int idx = tid; idx < BUF_TOTAL; idx += 1024) buf[idx] = 0.f;

    int lane = tid & 31;
    int wave = tid >> 5;        // 0..31
    int col  = lane & 15;       // timestep column within tile / A row index
    int half = lane >> 4;       // 0 or 1

    // A operand: loop-invariant, kept in VGPRs for entire run.
    // 16x4 f32 A layout: lanes 0-15 M=0..15, VGPR0 K=0 VGPR1 K=1; lanes 16-31 K=2,3.
    const float* Agb = Ag + (size_t)b * 256;
    v2f a[4];
    #pragma unroll
    for (int kb = 0; kb < 4; ++kb) {
        int kk = 4 * kb + 2 * half;
        a[kb].x = Agb[col * 16 + kk];
        a[kb].y = Agb[col * 16 + kk + 1];
    }
    // read-line metadata (this lane's B-operand rows): i = 4*kb + 2*half + rr
    float cg[8];
    int rbase[8], rdel[8], rslot[8];
    #pragma unroll
    for (int kb = 0; kb < 4; ++kb) {
        #pragma unroll
        for (int rr = 0; rr < 2; ++rr) {
            int jx = 2 * kb + rr;
            int li = 4 * kb + 2 * half + rr;
            cg[jx]    = Cv[b * 16 + li];
            rbase[jx] = c_off[li];
            rdel[jx]  = c_delays[li];
        }
    }
    // write-line metadata (this lane's D rows): i = r + 8*half
    float bvreg[8];
    int wbase[8], wdel[8], wslot[8];
    #pragma unroll
    for (int r = 0; r < 8; ++r) {
        int li = r + 8 * half;
        bvreg[r] = Bv[b * 16 + li];
        wbase[r] = c_off[li];
        wdel[r]  = c_delays[li];
    }
    // circular slots for first tile (t0 = wave); afterwards advance by 512 steps/iter
    int n = 16 * wave + col;
    #pragma unroll
    for (int jx = 0; jx < 8; ++jx) rslot[jx] = n % rdel[jx];
    #pragma unroll
    for (int r = 0; r < 8; ++r)   wslot[r] = n % wdel[r];

    __syncthreads();

    // 3000 tiles total; wave handles tile (wave + 32*k). 94 chunks of 512 steps
    // (512 <= min delay 1009 => all steps in a chunk are independent).
    const int ITERS = 94;
    for (int k = 0; k < ITERS; ++k) {
        if (wave + 32 * k < 3000) {   // wave-uniform: EXEC stays all-1s for WMMA
            // gather B operand from LDS delay lines + accumulate output partial
            v2f bm[4];
            float part = 0.f;
            #pragma unroll
            for (int kb = 0; kb < 4; ++kb) {
                float b0 = buf[rbase[2 * kb]     + rslot[2 * kb]];
                float b1 = buf[rbase[2 * kb + 1] + rslot[2 * kb + 1]];
                bm[kb].x = b0; bm[kb].y = b1;
                part += cg[2 * kb] * b0 + cg[2 * kb + 1] * b1;
            }
            // Nxt[16 x 16 timesteps] = A_g(16x16) @ D(16x16): 4 chained f32 WMMAs (K=4)
            v8f acc = {0.f, 0.f, 0.f, 0.f, 0.f, 0.f, 0.f, 0.f};
            acc = __builtin_amdgcn_wmma_f32_16x16x4_f32(false, a[0], false, bm[0], (short)0, acc, false, false);
            acc = __builtin_amdgcn_wmma_f32_16x16x4_f32(false, a[1], false, bm[1], (short)0, acc, false, false);
            acc = __builtin_amdgcn_wmma_f32_16x16x4_f32(false, a[2], false, bm[2], (short)0, acc, false, false);
            acc = __builtin_amdgcn_wmma_f32_16x16x4_f32(false, a[3], false, bm[3], (short)0, acc, false, false);

            // y[n] = Cv . d(n): combine the two half-wave partials
            float other = __shfl_xor(part, 16, 32);
            if (half == 0) y[(size_t)b * IR_LEN + n] = part + other;

            // impulse injection at n == 0 (lanes 0 and 16 of wave 0, first iteration)
            if (n == 0) {
                #pragma unroll
                for (int r = 0; r < 8; ++r) acc[r] += bvreg[r];
            }
            // write next delay-line inputs back at the just-read circular slots
            #pragma unroll
            for (int r = 0; r < 8; ++r) buf[wbase[r] + wslot[r]] = acc[r];

            // advance 512 steps (stride between this wave's consecutive tiles)
            #pragma unroll
            for (int jx = 0; jx < 8; ++jx) { rslot[jx] += 512; if (rslot[jx] >= rdel[jx]) rslot[jx] -= rdel[jx]; }
            #pragma unroll
            for (int r = 0; r < 8; ++r)   { wslot[r] += 512; if (wslot[r] >= wdel[r]) wslot[r] -= wdel[r]; }
            n += 512;
        }
        __syncthreads();   // chunk boundary: writes of chunk k visible to chunk k+1
    }
}

extern "C" void kernel_launch(void* const* d_in, const int* in_sizes, int n_in,
                              void* d_out, int out_size, void* d_ws, size_t ws_size,
                              hipStream_t stream) {
    const float* x  = (const float*)d_in[0];
    const float* WA = (const float*)d_in[1];
    const float* bA = (const float*)d_in[2];
    const float* WB = (const float*)d_in[3];
    const float* bB = (const float*)d_in[4];
    const float* WC = (const float*)d_in[5];
    const float* bC = (const float*)d_in[6];
    float* y  = (float*)d_out;
    float* ws = (float*)d_ws;

    float* fpart = ws;                        // 32*4*256 = 32768
    float* Ag    = ws + 32768;                // 32*256   =  8192
    float* Bvw   = ws + 40960;                // 32*16
    float* Cvw   = ws + 41472;                // 32*16

    feat_kernel<<<dim3(BATCH, FSPLIT), FEAT, 0, stream>>>(x, fpart);
    head_expm_kernel<<<BATCH, 256, 0, stream>>>(fpart, WA, bA, WB, bB, WC, bC, Ag, Bvw, Cvw);
    fdn_kernel<<<BATCH, 1024, 0, stream>>>(Ag, Bvw, Cvw, y);
}